// PointNet_SA_Module_KNN_1967095021880
// MI455X (gfx1250) — compile-verified
//
#include <hip/hip_runtime.h>

// ---------------- problem constants ----------------
constexpr int BATCH  = 8;
constexpr int NPTS   = 8192;    // N
constexpr int FIN    = 64;
constexpr int NPOINT = 2048;
constexpr int KNN    = 16;
constexpr int TC     = BATCH * NPOINT * KNN; // 262144 columns
constexpr float EPS  = 1e-5f;

typedef __attribute__((ext_vector_type(2))) float v2f;
typedef __attribute__((ext_vector_type(8))) float v8f;

// =====================================================================
// 1) Furthest point sampling: one block per batch, 8 pts per thread in
//    registers, wave-shuffle argmax + 32-entry LDS tree. Writes new_xyz.
// =====================================================================
__global__ __launch_bounds__(1024)
void fps_kernel(const float* __restrict__ xyz, float* __restrict__ out_xyz) {
  const int b = blockIdx.x;
  const int t = threadIdx.x;            // 0..1023
  const float* xb = xyz + (size_t)b * 3 * NPTS;
  float* ob = out_xyz + (size_t)b * 3 * NPOINT;

  float px[8], py[8], pz[8], dd[8];
#pragma unroll
  for (int i = 0; i < 8; ++i) {
    int n = t + (i << 10);
    px[i] = xb[n];
    py[i] = xb[NPTS + n];
    pz[i] = xb[2 * NPTS + n];
    dd[i] = 1e10f;
  }

  __shared__ float wD[32];
  __shared__ int   wI[32];
  __shared__ int   bl;

  int last = 0;
  if (t == 0) {  // first sample is index 0
    ob[0]          = xb[0];
    ob[NPOINT]     = xb[NPTS];
    ob[2 * NPOINT] = xb[2 * NPTS];
  }

  for (int s = 1; s < NPOINT; ++s) {
    // coords of last selected point (uniform -> cached/broadcast load)
    float cx = xb[last], cy = xb[NPTS + last], cz = xb[2 * NPTS + last];

    float bestd = -1.0f; int bestn = 0;
#pragma unroll
    for (int i = 0; i < 8; ++i) {
      float dx = px[i] - cx, dy = py[i] - cy, dz = pz[i] - cz;
      float d = dx * dx + dy * dy + dz * dz;
      dd[i] = fminf(dd[i], d);
      int n = t + (i << 10);
      if (dd[i] > bestd) { bestd = dd[i]; bestn = n; }  // ascending n: '>' keeps first
    }
    // intra-wave reduce (argmax, tie -> smaller index)
#pragma unroll
    for (int m = 1; m < 32; m <<= 1) {
      float od = __shfl_xor(bestd, m, 32);
      int   on = __shfl_xor(bestn, m, 32);
      if (od > bestd || (od == bestd && on < bestn)) { bestd = od; bestn = on; }
    }
    if ((t & 31) == 0) { wD[t >> 5] = bestd; wI[t >> 5] = bestn; }
    __syncthreads();
    if (t < 32) {
      bestd = wD[t]; bestn = wI[t];
#pragma unroll
      for (int m = 1; m < 32; m <<= 1) {
        float od = __shfl_xor(bestd, m, 32);
        int   on = __shfl_xor(bestn, m, 32);
        if (od > bestd || (od == bestd && on < bestn)) { bestd = od; bestn = on; }
      }
      if (t == 0) {
        bl = bestn;
        ob[s]              = xb[bestn];
        ob[NPOINT + s]     = xb[NPTS + bestn];
        ob[2 * NPOINT + s] = xb[2 * NPTS + bestn];
      }
    }
    __syncthreads();
    last = bl;
  }
}

// =====================================================================
// 2) KNN: one thread per query, candidates tiled through LDS,
//    register-resident top-16 via worst-slot replacement (fully unrolled
//    so the arrays stay in VGPRs). Neighbor ORDER is irrelevant: it only
//    feeds order-invariant BN stats and a max-pool.
// =====================================================================
constexpr int KTS = 512;
__global__ __launch_bounds__(256)
void knn_kernel(const float* __restrict__ xyz, const float* __restrict__ newxyz,
                int* __restrict__ knn_idx) {
  const int q = blockIdx.x * 256 + threadIdx.x;   // 16384 queries
  const int b = q >> 11;                          // NPOINT = 2048
  const int p = q & (NPOINT - 1);
  const float* xb = xyz + (size_t)b * 3 * NPTS;
  const float qx = newxyz[(size_t)b * 3 * NPOINT + p];
  const float qy = newxyz[(size_t)b * 3 * NPOINT + NPOINT + p];
  const float qz = newxyz[(size_t)b * 3 * NPOINT + 2 * NPOINT + p];

  __shared__ float xs[KTS], ys[KTS], zs[KTS];

  float kd[16]; int ki[16];
#pragma unroll
  for (int i = 0; i < 16; ++i) { kd[i] = 1e30f; ki[i] = 0; }
  float worst = 1e30f; int wslot = 15;

  for (int nb = 0; nb < NPTS; nb += KTS) {
    __syncthreads();
    for (int j = threadIdx.x; j < KTS; j += 256) {
      xs[j] = xb[nb + j];
      ys[j] = xb[NPTS + nb + j];
      zs[j] = xb[2 * NPTS + nb + j];
    }
    __syncthreads();
    for (int j = 0; j < KTS; ++j) {
      float dx = qx - xs[j], dy = qy - ys[j], dz = qz - zs[j];
      float d = dx * dx + dy * dy + dz * dz;
      if (d < worst) {
        int n = nb + j;
#pragma unroll
        for (int s2 = 0; s2 < 16; ++s2) if (s2 == wslot) { kd[s2] = d; ki[s2] = n; }
        worst = kd[0]; wslot = 0;
#pragma unroll
        for (int s2 = 1; s2 < 16; ++s2) if (kd[s2] > worst) { worst = kd[s2]; wslot = s2; }
      }
    }
  }
  int* o = knn_idx + (size_t)q * KNN;
#pragma unroll
  for (int i = 0; i < 16; ++i) o[i] = ki[i];
}

// =====================================================================
// 3) GEMM1: y1[64 x TC] = W1[64x67] @ gathered X[67 x TC] + b1
//    f32 WMMA 16x16x4. 128 threads, 64-col tiles, K padded 67->68.
//    LDS tile stored TRANSPOSED (XsT[col][k], k-stride 68) so each
//    B fragment is one aligned contiguous ds_load_b64 (no repack movs).
// =====================================================================
__global__ __launch_bounds__(128)
void gemm1_kernel(const float* __restrict__ xyz, const float* __restrict__ points,
                  const float* __restrict__ W1, const float* __restrict__ b1,
                  const float* __restrict__ newxyz, const int* __restrict__ knn_idx,
                  float* __restrict__ y1) {
  __shared__ float XsT[64][68];   // [col][k], 68 = 67 + zero pad row, even stride
  __shared__ int ncol[64];
  __shared__ int pcol[64];
  const int tid = threadIdx.x;
  const int colBase = blockIdx.x * 64;
  const int bidx = colBase >> 15;              // NPOINT*KNN = 32768

  if (tid < 64) {
    int gcol = colBase + tid;
    ncol[tid] = knn_idx[gcol];
    pcol[tid] = (gcol & 32767) >> 4;
  }
  __syncthreads();

  for (int e = tid; e < 68 * 64; e += 128) {
    int row = e >> 6, col = e & 63;            // row = k index 0..67
    float v = 0.0f;
    if (row < 3) {
      v = xyz[((size_t)bidx * 3 + row) * NPTS + ncol[col]]
        - newxyz[((size_t)bidx * 3 + row) * NPOINT + pcol[col]];
    } else if (row < 67) {
      v = points[((size_t)bidx * FIN + (row - 3)) * NPTS + ncol[col]];
    }
    XsT[col][row] = v;
  }
  __syncthreads();

  const int lane = tid & 31;
  const int w    = tid >> 5;            // wave 0..3 -> 16-row strip
  const int lm   = lane & 15;
  const int kb   = (lane < 16) ? 0 : 2; // A/B K-half per ISA layout
  const int rowA = w * 16 + lm;

  v8f zero = {0.f,0.f,0.f,0.f,0.f,0.f,0.f,0.f};
  v8f acc[4] = {zero, zero, zero, zero};

  for (int k0 = 0; k0 < 68; k0 += 4) {
    int ka = k0 + kb;                   // even -> 8B-aligned LDS pairs
    v2f a;
    a.x = (ka     < 67) ? W1[rowA * 67 + ka]     : 0.0f;
    a.y = (ka + 1 < 67) ? W1[rowA * 67 + ka + 1] : 0.0f;
#pragma unroll
    for (int t2 = 0; t2 < 4; ++t2) {
      v2f bb = *(const v2f*)(&XsT[t2 * 16 + lm][ka]);
      acc[t2] = __builtin_amdgcn_wmma_f32_16x16x4_f32(
          false, a, false, bb, (short)0, acc[t2], false, false);
    }
  }

  const int rhi = (lane < 16) ? 0 : 8;
#pragma unroll
  for (int r = 0; r < 8; ++r) {
    int row = w * 16 + r + rhi;
    float bv = b1[row];
#pragma unroll
    for (int t2 = 0; t2 < 4; ++t2)
      y1[(size_t)row * TC + colBase + t2 * 16 + lm] = acc[t2][r] + bv;
  }
}

// =====================================================================
// 4) BN stats: one block per channel, deterministic strided + tree
//    reduction; emits fused scale/shift so BN+ReLU is a single FMA later.
// =====================================================================
__global__ __launch_bounds__(256)
void stats_kernel(const float* __restrict__ y, const float* __restrict__ g,
                  const float* __restrict__ be, float* __restrict__ scale,
                  float* __restrict__ shift) {
  const int ch = blockIdx.x;
  const int t  = threadIdx.x;
  const float* row = y + (size_t)ch * TC;
  float s = 0.f, sq = 0.f;
  for (int j = t; j < TC; j += 256) { float v = row[j]; s += v; sq += v * v; }
  __shared__ float ss[256], sqs[256];
  ss[t] = s; sqs[t] = sq;
  __syncthreads();
  for (int off = 128; off > 0; off >>= 1) {
    if (t < off) { ss[t] += ss[t + off]; sqs[t] += sqs[t + off]; }
    __syncthreads();
  }
  if (t == 0) {
    float mean = ss[0] / (float)TC;
    float var  = sqs[0] / (float)TC - mean * mean;
    float sc   = g[ch] * rsqrtf(var + EPS);
    scale[ch] = sc;
    shift[ch] = be[ch] - mean * sc;
  }
}

// =====================================================================
// 5) GEMM2: y2 = W2[64x64] @ relu(BN1(y1)) + b2   (BN fused on LDS load)
// =====================================================================
__global__ __launch_bounds__(128)
void gemm2_kernel(const float* __restrict__ y1, const float* __restrict__ W2,
                  const float* __restrict__ b2, const float* __restrict__ scale1,
                  const float* __restrict__ shift1, float* __restrict__ y2) {
  __shared__ float XsT[64][68];   // [col][k], stride padded to 68
  const int tid = threadIdx.x;
  const int colBase = blockIdx.x * 64;

  for (int e = tid; e < 64 * 64; e += 128) {
    int row = e >> 6, col = e & 63;            // row = k channel
    float v = y1[(size_t)row * TC + colBase + col] * scale1[row] + shift1[row];
    XsT[col][row] = fmaxf(v, 0.0f);
  }
  __syncthreads();

  const int lane = tid & 31;
  const int w    = tid >> 5;
  const int lm   = lane & 15;
  const int kb   = (lane < 16) ? 0 : 2;
  const int rowA = w * 16 + lm;

  v8f zero = {0.f,0.f,0.f,0.f,0.f,0.f,0.f,0.f};
  v8f acc[4] = {zero, zero, zero, zero};

  for (int k0 = 0; k0 < 64; k0 += 4) {
    int ka = k0 + kb;
    v2f a;
    a.x = W2[rowA * 64 + ka];
    a.y = W2[rowA * 64 + ka + 1];
#pragma unroll
    for (int t2 = 0; t2 < 4; ++t2) {
      v2f bb = *(const v2f*)(&XsT[t2 * 16 + lm][ka]);
      acc[t2] = __builtin_amdgcn_wmma_f32_16x16x4_f32(
          false, a, false, bb, (short)0, acc[t2], false, false);
    }
  }

  const int rhi = (lane < 16) ? 0 : 8;
#pragma unroll
  for (int r = 0; r < 8; ++r) {
    int row = w * 16 + r + rhi;
    float bv = b2[row];
#pragma unroll
    for (int t2 = 0; t2 < 4; ++t2)
      y2[(size_t)row * TC + colBase + t2 * 16 + lm] = acc[t2][r] + bv;
  }
}

// =====================================================================
// 6) GEMM3 + maxpool: y3 = W3[128x64] @ relu(BN2(y2)) + b3, then max over
//    K=16 neighbors == one 16-col WMMA tile -> shfl_xor max within each 16-lane half.
// =====================================================================
__global__ __launch_bounds__(128)
void gemm3_kernel(const float* __restrict__ y2, const float* __restrict__ W3,
                  const float* __restrict__ b3, const float* __restrict__ scale2,
                  const float* __restrict__ shift2, float* __restrict__ out_feat) {
  __shared__ float XsT[64][68];   // [col][k], stride padded to 68
  const int tid = threadIdx.x;
  const int colBase = blockIdx.x * 64;

  for (int e = tid; e < 64 * 64; e += 128) {
    int row = e >> 6, col = e & 63;
    float v = y2[(size_t)row * TC + colBase + col] * scale2[row] + shift2[row];
    XsT[col][row] = fmaxf(v, 0.0f);
  }
  __syncthreads();

  const int lane = tid & 31;
  const int w    = tid >> 5;
  const int lm   = lane & 15;
  const int kb   = (lane < 16) ? 0 : 2;
  const int rA   = w * 16 + lm;        // rows 0..63
  const int rB   = 64 + w * 16 + lm;   // rows 64..127

  v8f zero = {0.f,0.f,0.f,0.f,0.f,0.f,0.f,0.f};
  v8f accA[4] = {zero, zero, zero, zero};
  v8f accB[4] = {zero, zero, zero, zero};

  for (int k0 = 0; k0 < 64; k0 += 4) {
    int ka = k0 + kb;
    v2f a0, a1;
    a0.x = W3[rA * 64 + ka];  a0.y = W3[rA * 64 + ka + 1];
    a1.x = W3[rB * 64 + ka];  a1.y = W3[rB * 64 + ka + 1];
#pragma unroll
    for (int t2 = 0; t2 < 4; ++t2) {
      v2f bb = *(const v2f*)(&XsT[t2 * 16 + lm][ka]);
      accA[t2] = __builtin_amdgcn_wmma_f32_16x16x4_f32(
          false, a0, false, bb, (short)0, accA[t2], false, false);
      accB[t2] = __builtin_amdgcn_wmma_f32_16x16x4_f32(
          false, a1, false, bb, (short)0, accB[t2], false, false);
    }
  }

  const int rhi = (lane < 16) ? 0 : 8;
#pragma unroll
  for (int t2 = 0; t2 < 4; ++t2) {
    int gcol = colBase + t2 * 16;          // first col of this (b,p) group
    int bb2  = gcol >> 15;
    int p    = (gcol & 32767) >> 4;
    float va[8], vb[8];
#pragma unroll
    for (int r = 0; r < 8; ++r) {
      va[r] = accA[t2][r] + b3[w * 16 + r + rhi];
      vb[r] = accB[t2][r] + b3[64 + w * 16 + r + rhi];
    }
#pragma unroll
    for (int m = 1; m < 16; m <<= 1) {
#pragma unroll
      for (int r = 0; r < 8; ++r) {
        va[r] = fmaxf(va[r], __shfl_xor(va[r], m, 32));
        vb[r] = fmaxf(vb[r], __shfl_xor(vb[r], m, 32));
      }
    }
    if (lm == 0) {
#pragma unroll
      for (int r = 0; r < 8; ++r) {
        out_feat[((size_t)bb2 * 128 + w * 16 + r + rhi) * NPOINT + p]      = va[r];
        out_feat[((size_t)bb2 * 128 + 64 + w * 16 + r + rhi) * NPOINT + p] = vb[r];
      }
    }
  }
}

// =====================================================================
// launch
// =====================================================================
extern "C" void kernel_launch(void* const* d_in, const int* in_sizes, int n_in,
                              void* d_out, int out_size, void* d_ws, size_t ws_size,
                              hipStream_t stream) {
  (void)in_sizes; (void)n_in; (void)out_size; (void)ws_size;
  const float* xyz    = (const float*)d_in[0];
  const float* points = (const float*)d_in[1];
  const float* W1 = (const float*)d_in[2];  const float* b1 = (const float*)d_in[3];
  const float* g1 = (const float*)d_in[4];  const float* be1 = (const float*)d_in[5];
  const float* W2 = (const float*)d_in[6];  const float* b2 = (const float*)d_in[7];
  const float* g2 = (const float*)d_in[8];  const float* be2 = (const float*)d_in[9];
  const float* W3 = (const float*)d_in[10]; const float* b3 = (const float*)d_in[11];

  float* out     = (float*)d_out;
  float* newxyz  = out;                               // (B,3,NPOINT)
  float* newfeat = out + (size_t)BATCH * 3 * NPOINT;  // (B,128,NPOINT)

  char* ws = (char*)d_ws;
  int*   knn_idx = (int*)ws;                              // 1 MB
  float* scale1  = (float*)(ws + (1 << 20));
  float* shift1  = scale1 + 64;
  float* scale2  = shift1 + 64;
  float* shift2  = scale2 + 64;
  float* y1      = (float*)(ws + (2 << 20));              // 64*TC f32 = 67 MB
  float* y2      = y1 + (size_t)64 * TC;                  // 67 MB

  fps_kernel<<<BATCH, 1024, 0, stream>>>(xyz, newxyz);
  knn_kernel<<<(BATCH * NPOINT) / 256, 256, 0, stream>>>(xyz, newxyz, knn_idx);
  gemm1_kernel<<<TC / 64, 128, 0, stream>>>(xyz, points, W1, b1, newxyz, knn_idx, y1);
  stats_kernel<<<64, 256, 0, stream>>>(y1, g1, be1, scale1, shift1);
  gemm2_kernel<<<TC / 64, 128, 0, stream>>>(y1, W2, b2, scale1, shift1, y2);
  stats_kernel<<<64, 256, 0, stream>>>(y2, g2, be2, scale2, shift2);
  gemm3_kernel<<<TC / 64, 128, 0, stream>>>(y2, W3, b3, scale2, shift2, newfeat);
}